// MyLinear_59674275611289
// MI455X (gfx1250) — compile-verified
//
#include <hip/hip_runtime.h>
#include <hip/hip_bf16.h>
#include <math.h>

// Problem sizes (fixed by the reference)
#define IN_U   2048
#define OUT_U  2048
#define HID    10
#define BATCH  4096

typedef __attribute__((ext_vector_type(2))) float fx2;
typedef __attribute__((ext_vector_type(8))) float fx8;
typedef __attribute__((ext_vector_type(4))) int   i32x4;

// ---- CDNA5 async global->LDS path (guarded; falls back to sync copies) ----
#if __has_builtin(__builtin_amdgcn_global_load_async_to_lds_b128)
#define HAVE_ASYNC_LDS 1
typedef __attribute__((address_space(1))) i32x4 g_i32x4;  // global AS
typedef __attribute__((address_space(3))) i32x4 l_i32x4;  // LDS AS
#else
#define HAVE_ASYNC_LDS 0
#endif

#if __has_builtin(__builtin_amdgcn_s_wait_asynccnt)
#define WAIT_ASYNC(n) __builtin_amdgcn_s_wait_asynccnt(n)
#else
#define WAIT_ASYNC(n) asm volatile("s_wait_asynccnt %0" ::"i"(n) : "memory")
#endif

__device__ __forceinline__ void cp16_to_lds(void* lds_dst, const void* gsrc) {
#if HAVE_ASYNC_LDS
  __builtin_amdgcn_global_load_async_to_lds_b128(
      (g_i32x4*)gsrc, (l_i32x4*)lds_dst, /*offset=*/0, /*cpol=*/0);
#else
  *(float4*)lds_dst = *(const float4*)gsrc;
#endif
}

// ---------------------------------------------------------------------------
// Kernel 1: A[i,h] = sum_k W[i,k] * W1[k,h]        (bid <  2048)
//           B[j,h] = sum_k W[k,j] * W1[2048+k,h]   (bid >= 2048)
// ---------------------------------------------------------------------------
__global__ __launch_bounds__(256) void compute_AB(const float* __restrict__ W,
                                                  const float* __restrict__ W1,
                                                  float* __restrict__ Aout,
                                                  float* __restrict__ Bout) {
  __shared__ float s[256 * HID];
  const int t = threadIdx.x;
  const int bid = blockIdx.x;

  float acc[HID];
#pragma unroll
  for (int h = 0; h < HID; ++h) acc[h] = 0.0f;

  if (bid < IN_U) {
    const int i = bid;
    for (int k = t; k < OUT_U; k += 256) {
      const float w = W[(size_t)i * OUT_U + k];
      const float* w1r = &W1[(size_t)k * HID];
#pragma unroll
      for (int h = 0; h < HID; ++h) acc[h] += w * w1r[h];
    }
  } else {
    const int j = bid - IN_U;
    for (int k = t; k < IN_U; k += 256) {
      const float w = W[(size_t)k * OUT_U + j];
      const float* w1r = &W1[(size_t)(OUT_U + k) * HID];
#pragma unroll
      for (int h = 0; h < HID; ++h) acc[h] += w * w1r[h];
    }
  }

#pragma unroll
  for (int h = 0; h < HID; ++h) s[t * HID + h] = acc[h];
  __syncthreads();
  for (int off = 128; off > 0; off >>= 1) {
    if (t < off) {
#pragma unroll
      for (int h = 0; h < HID; ++h) s[t * HID + h] += s[(t + off) * HID + h];
    }
    __syncthreads();
  }
  if (t == 0) {
    float* dst = (bid < IN_U) ? &Aout[(size_t)bid * HID]
                              : &Bout[(size_t)(bid - IN_U) * HID];
#pragma unroll
    for (int h = 0; h < HID; ++h) dst[h] = s[h];
  }
}

// ---------------------------------------------------------------------------
// Kernel 2: per-(i,j) tiny MLP -> W_new = W + delta
// ---------------------------------------------------------------------------
__global__ __launch_bounds__(256) void nca_wnew(const float* __restrict__ W,
                                                const float* __restrict__ Aw,
                                                const float* __restrict__ Bw,
                                                const float* __restrict__ b1p,
                                                const float* __restrict__ W2p,
                                                const float* __restrict__ b2p,
                                                const float* __restrict__ W3p,
                                                const float* __restrict__ b3p,
                                                float* __restrict__ Wnew) {
  __shared__ float sW2[HID * HID];
  __shared__ float sb1[HID], sb2[HID], sW3[HID];
  __shared__ float sb3;
  const int t = threadIdx.x;
  if (t < 100)       sW2[t]       = W2p[t];
  else if (t < 110)  sb1[t - 100] = b1p[t - 100];
  else if (t < 120)  sb2[t - 110] = b2p[t - 110];
  else if (t < 130)  sW3[t - 120] = W3p[t - 120];
  else if (t == 130) sb3          = b3p[0];
  __syncthreads();

  const int i = blockIdx.y;
  const int j = blockIdx.x * 256 + t;

  const float* a = &Aw[(size_t)i * HID];
  const float* b = &Bw[(size_t)j * HID];

  float h1[HID];
#pragma unroll
  for (int h = 0; h < HID; ++h) {
    const float v = a[h] + b[h] + sb1[h];
    h1[h] = v > 0.0f ? v : 0.0f;
  }
  float h2[HID];
#pragma unroll
  for (int g = 0; g < HID; ++g) h2[g] = sb2[g];
#pragma unroll
  for (int h = 0; h < HID; ++h) {
    const float v = h1[h];
#pragma unroll
    for (int g = 0; g < HID; ++g) h2[g] = fmaf(v, sW2[h * HID + g], h2[g]);
  }
  float delta = sb3;
#pragma unroll
  for (int g = 0; g < HID; ++g) {
    const float v = h2[g] > 0.0f ? h2[g] : 0.0f;
    delta = fmaf(v, sW3[g], delta);
  }
  const size_t idx = (size_t)i * OUT_U + j;
  Wnew[idx] = W[idx] + delta;
}

// ---------------------------------------------------------------------------
// Kernel 3: logits = X @ W_new with V_WMMA_F32_16X16X4_F32
// Block tile 128(M) x 64(N), K chunks of 32, double-buffered LDS filled with
// GLOBAL_LOAD_ASYNC_TO_LDS_B128 (ASYNCcnt), 8 waves each owning a 32x32 tile
// (4 accumulators -> each LDS fragment feeds 2 WMMAs, 4 independent chains).
// ---------------------------------------------------------------------------
#define TM 128
#define TN 64
#define TK 32
#define NCHUNK (OUT_U / TK)
#define APT 4   // X-tile float4 loads per thread per chunk (128*32/4/256)
#define BPT 2   // W-tile float4 loads per thread per chunk (32*64/4/256)

__global__ __launch_bounds__(256) void gemm_wmma(const float* __restrict__ X,
                                                 const float* __restrict__ Wn,
                                                 float* __restrict__ out) {
  __shared__ float Xs[2][TM][TK + 4];  // row stride 36 floats (bank-safe)
  __shared__ float Ws[2][TK][TN + 4];  // row stride 68 floats

  const int tid  = threadIdx.x;
  const int lane = tid & 31;
  const int wave = tid >> 5;
  const int half = lane >> 4;
  const int l16  = lane & 15;
  const int m0 = (wave & 3) * 32;   // wave's 32-row strip
  const int n0 = (wave >> 2) * 32;  // wave's 32-col strip
  const int blockM = blockIdx.y * TM;
  const int blockN = blockIdx.x * TN;

  fx8 acc00 = {}, acc01 = {}, acc10 = {}, acc11 = {};

  auto issue_tiles = [&](int buf, int k0) {
#pragma unroll
    for (int it = 0; it < APT; ++it) {
      const int f = (tid + it * 256) * 4;
      const int row = f >> 5;   // /32
      const int col = f & 31;
      cp16_to_lds(&Xs[buf][row][col],
                  &X[(size_t)(blockM + row) * OUT_U + k0 + col]);
    }
#pragma unroll
    for (int it = 0; it < BPT; ++it) {
      const int f = (tid + it * 256) * 4;
      const int row = f >> 6;   // /64
      const int col = f & 63;
      cp16_to_lds(&Ws[buf][row][col],
                  &Wn[(size_t)(k0 + row) * OUT_U + blockN + col]);
    }
  };

  issue_tiles(0, 0);

  for (int c = 0; c < NCHUNK; ++c) {
    const int cur = c & 1;
    if (c + 1 < NCHUNK) issue_tiles(cur ^ 1, (c + 1) * TK);
#if HAVE_ASYNC_LDS
    if (c + 1 < NCHUNK) {
      WAIT_ASYNC(APT + BPT);  // only next chunk's loads may remain in flight
    } else {
      WAIT_ASYNC(0);
    }
#endif
    __syncthreads();  // chunk c resident in Xs/Ws[cur] for all waves

#pragma unroll
    for (int kk = 0; kk < TK; kk += 4) {
      fx2 a0, a1, b0, b1;
      // A 16x4 fragment: vgpr0 = K(0|2), vgpr1 = K(1|3) by half-wave
      a0.x = Xs[cur][m0 + l16][kk + 2 * half + 0];
      a0.y = Xs[cur][m0 + l16][kk + 2 * half + 1];
      a1.x = Xs[cur][m0 + 16 + l16][kk + 2 * half + 0];
      a1.y = Xs[cur][m0 + 16 + l16][kk + 2 * half + 1];
      // B 4x16 fragment: vgpr0 = row K(0|2), vgpr1 = row K(1|3), N = l16
      b0.x = Ws[cur][kk + 2 * half + 0][n0 + l16];
      b0.y = Ws[cur][kk + 2 * half + 1][n0 + l16];
      b1.x = Ws[cur][kk + 2 * half + 0][n0 + 16 + l16];
      b1.y = Ws[cur][kk + 2 * half + 1][n0 + 16 + l16];
      acc00 = __builtin_amdgcn_wmma_f32_16x16x4_f32(false, a0, false, b0,
                                                    (short)0, acc00, false, false);
      acc01 = __builtin_amdgcn_wmma_f32_16x16x4_f32(false, a0, false, b1,
                                                    (short)0, acc01, false, false);
      acc10 = __builtin_amdgcn_wmma_f32_16x16x4_f32(false, a1, false, b0,
                                                    (short)0, acc10, false, false);
      acc11 = __builtin_amdgcn_wmma_f32_16x16x4_f32(false, a1, false, b1,
                                                    (short)0, acc11, false, false);
    }
    __syncthreads();  // protect buf reuse before next issue
  }

  // C/D layout: VGPR r -> M = r + half*8, N = l16
#pragma unroll
  for (int r = 0; r < 8; ++r) {
    const int gm0 = blockM + m0 + r + half * 8;
    const int gm1 = gm0 + 16;
    const int gn0 = blockN + n0 + l16;
    out[(size_t)gm0 * OUT_U + gn0]      = acc00[r];
    out[(size_t)gm0 * OUT_U + gn0 + 16] = acc01[r];
    out[(size_t)gm1 * OUT_U + gn0]      = acc10[r];
    out[(size_t)gm1 * OUT_U + gn0 + 16] = acc11[r];
  }
}

// ---------------------------------------------------------------------------
// Kernel 4: in-place numerically-stable row softmax over d_out (4096 x 2048)
// ---------------------------------------------------------------------------
__global__ __launch_bounds__(256) void softmax_rows(float* __restrict__ out) {
  __shared__ float red[256];
  const int t = threadIdx.x;
  float* row = out + (size_t)blockIdx.x * OUT_U;

  float v[8];
  float mx = -INFINITY;
#pragma unroll
  for (int s = 0; s < 8; ++s) {
    v[s] = row[t + s * 256];
    mx = fmaxf(mx, v[s]);
  }
  red[t] = mx;
  __syncthreads();
  for (int off = 128; off > 0; off >>= 1) {
    if (t < off) red[t] = fmaxf(red[t], red[t + off]);
    __syncthreads();
  }
  mx = red[0];
  __syncthreads();

  float sum = 0.0f;
#pragma unroll
  for (int s = 0; s < 8; ++s) {
    v[s] = __expf(v[s] - mx);
    sum += v[s];
  }
  red[t] = sum;
  __syncthreads();
  for (int off = 128; off > 0; off >>= 1) {
    if (t < off) red[t] += red[t + off];
    __syncthreads();
  }
  const float inv = 1.0f / red[0];
#pragma unroll
  for (int s = 0; s < 8; ++s) row[t + s * 256] = v[s] * inv;
}

// ---------------------------------------------------------------------------
// Launch: ws layout = A[2048*10] | B[2048*10] | W_new[2048*2048]  (~16.2 MB)
// ---------------------------------------------------------------------------
extern "C" void kernel_launch(void* const* d_in, const int* in_sizes, int n_in,
                              void* d_out, int out_size, void* d_ws,
                              size_t ws_size, hipStream_t stream) {
  const float* X  = (const float*)d_in[0];
  const float* W  = (const float*)d_in[1];
  const float* W1 = (const float*)d_in[2];
  const float* b1 = (const float*)d_in[3];
  const float* W2 = (const float*)d_in[4];
  const float* b2 = (const float*)d_in[5];
  const float* W3 = (const float*)d_in[6];
  const float* b3 = (const float*)d_in[7];
  float* out = (float*)d_out;

  float* ws = (float*)d_ws;
  float* Aw = ws;                            // 2048*10
  float* Bw = ws + (size_t)IN_U * HID;       // 2048*10
  float* Wn = ws + (size_t)2 * IN_U * HID;   // 2048*2048

  compute_AB<<<IN_U + OUT_U, 256, 0, stream>>>(W, W1, Aw, Bw);
  nca_wnew<<<dim3(OUT_U / 256, IN_U), 256, 0, stream>>>(W, Aw, Bw, b1, W2, b2,
                                                        W3, b3, Wn);
  gemm_wmma<<<dim3(OUT_U / TN, BATCH / TM), 256, 0, stream>>>(X, Wn, out);
  softmax_rows<<<BATCH, 256, 0, stream>>>(out);
}